// appnp_8589935067
// MI455X (gfx1250) — compile-verified
//
#include <hip/hip_runtime.h>
#include <math.h>

#define N_NODES 100000
#define E_EDGES 1600000
#define IN_CH   256
#define OUT_CH  64
#define K_STEPS 10
#define ALPHA   0.1f

typedef __attribute__((ext_vector_type(2))) float v2f;
typedef __attribute__((ext_vector_type(8))) float v8f;

// ---------------- degree / normalization ----------------

__global__ __launch_bounds__(256) void k_deg_init(float* __restrict__ deg) {
  int i = blockIdx.x * 256 + threadIdx.x;
  if (i < N_NODES) deg[i] = 1.0f;  // self-loop contribution
}

__global__ __launch_bounds__(256) void k_deg_count(const int* __restrict__ dst,
                                                   float* __restrict__ deg) {
  int e = blockIdx.x * 256 + threadIdx.x;
  if (e < E_EDGES) atomicAdd(&deg[dst[e]], 1.0f);
}

__global__ __launch_bounds__(256) void k_dinv(float* __restrict__ deg) {
  int i = blockIdx.x * 256 + threadIdx.x;
  if (i < N_NODES) deg[i] = rsqrtf(deg[i]);  // deg >= 1 always (self-loop)
}

// ---------------- h = relu(x @ W + b) via f32 WMMA ----------------
// One wave computes one 16x16 tile of h. 8 waves/block -> 2 M-tiles x 4 N-tiles.
// A (16x4 f32): lanes 0-15 = rows, VGPR0/1 = K {0,1}; lanes 16-31 = K {2,3}.
// B (4x16 f32): VGPR0 = row K0 (lanes 0-15) / K2 (lanes 16-31); VGPR1 = K1/K3.
// C/D (16x16 f32): VGPR v -> M=v (lanes 0-15), M=v+8 (lanes 16-31).

__global__ __launch_bounds__(256) void k_gemm_relu(const float* __restrict__ x,
                                                   const float* __restrict__ W,
                                                   const float* __restrict__ bias,
                                                   float* __restrict__ h,
                                                   float* __restrict__ z) {
  const int lane  = threadIdx.x & 31;
  const int wave  = threadIdx.x >> 5;          // 0..7
  const int mt    = blockIdx.x * 2 + (wave >> 2);
  const int nt    = wave & 3;
  const int m0    = mt * 16;
  const int n0    = nt * 16;
  const int lrow  = lane & 15;                 // A row / B-C column within tile
  const int khalf = lane >> 4;                 // 0: K{0,1}, 1: K{2,3}

  const float* __restrict__ arow = x + (size_t)(m0 + lrow) * IN_CH + 2 * khalf;
  const float* __restrict__ bcol = W + (size_t)(2 * khalf) * OUT_CH + (n0 + lrow);

  v8f c = {};
  #pragma unroll 4
  for (int k = 0; k < IN_CH; k += 4) {
    v2f a = *(const v2f*)(arow + k);           // contiguous K pair -> b64 load
    v2f bv;
    bv.x = bcol[(size_t)k * OUT_CH];
    bv.y = bcol[(size_t)(k + 1) * OUT_CH];
    // 8 args: (neg_a, A, neg_b, B, c_mod, C, reuse_a, reuse_b)
    c = __builtin_amdgcn_wmma_f32_16x16x4_f32(false, a, false, bv,
                                              (short)0, c, false, false);
  }

  const float bval  = bias[n0 + lrow];
  const int   mbase = m0 + 8 * khalf;
  #pragma unroll
  for (int v = 0; v < 8; ++v) {
    float val = c[v] + bval;
    val = val > 0.0f ? val : 0.0f;             // ReLU
    size_t idx = (size_t)(mbase + v) * OUT_CH + n0 + lrow;
    h[idx] = val;                              // keep h for the alpha*h term
    z[idx] = val;                              // z starts as h
  }
}

// ---------------- propagation step ----------------

__global__ __launch_bounds__(256) void k_zero(float* __restrict__ p) {
  int i = blockIdx.x * 256 + threadIdx.x;
  if (i < N_NODES * OUT_CH) p[i] = 0.0f;
}

// One wave per edge; each lane handles 2 channels (float2 gather, 2 f32 atomics).
// Edge id is wave-uniform -> src/dst/dinv loads are wave-coherent, L2-resident.
__global__ __launch_bounds__(256) void k_scatter(const int* __restrict__ src,
                                                 const int* __restrict__ dst,
                                                 const float* __restrict__ dinv,
                                                 const float* __restrict__ z,
                                                 float* __restrict__ acc) {
  const int e    = blockIdx.x * 8 + (threadIdx.x >> 5);   // 8 edges per block
  const int c2   = (threadIdx.x & 31) * 2;                // channel pair
  const int s    = src[e];
  const int d    = dst[e];
  const float w  = dinv[s] * dinv[d];
  v2f zv = *(const v2f*)(z + (size_t)s * OUT_CH + c2);
  float* out = acc + (size_t)d * OUT_CH + c2;
  atomicAdd(out + 0, w * zv.x);
  atomicAdd(out + 1, w * zv.y);
}

// z = (1-alpha) * (acc + dinv^2 * z) + alpha * h   (self-loop folded in here)
__global__ __launch_bounds__(256) void k_combine(const float* __restrict__ acc,
                                                 const float* __restrict__ h,
                                                 const float* __restrict__ dinv,
                                                 float* __restrict__ z) {
  int i = blockIdx.x * 256 + threadIdx.x;
  if (i >= N_NODES * OUT_CH) return;
  int node = i >> 6;  // OUT_CH == 64
  float di = dinv[node];
  float v  = acc[i] + di * di * z[i];
  z[i] = (1.0f - ALPHA) * v + ALPHA * h[i];
}

// ---------------- log-softmax (row-wise, wave32 per row) ----------------

__global__ __launch_bounds__(256) void k_logsoftmax(float* __restrict__ z) {
  const int row  = blockIdx.x * 8 + (threadIdx.x >> 5);
  const int lane = threadIdx.x & 31;
  if (row >= N_NODES) return;
  float* p = z + (size_t)row * OUT_CH;
  float a = p[lane];
  float b = p[lane + 32];
  float m = fmaxf(a, b);
  #pragma unroll
  for (int off = 16; off > 0; off >>= 1) m = fmaxf(m, __shfl_xor(m, off, 32));
  float s = expf(a - m) + expf(b - m);
  #pragma unroll
  for (int off = 16; off > 0; off >>= 1) s += __shfl_xor(s, off, 32);
  float lse = m + logf(s);
  p[lane]      = a - lse;
  p[lane + 32] = b - lse;
}

// ---------------- host ----------------

extern "C" void kernel_launch(void* const* d_in, const int* in_sizes, int n_in,
                              void* d_out, int out_size, void* d_ws, size_t ws_size,
                              hipStream_t stream) {
  const float* x          = (const float*)d_in[0];
  const int*   edge_index = (const int*)d_in[1];   // [2, E] row-major: row0=src, row1=dst
  // d_in[2] = edge_attr, unused by APPNP
  const float* W          = (const float*)d_in[3];
  const float* b          = (const float*)d_in[4];

  float* z   = (float*)d_out;                         // z lives in d_out (overwritten fully)
  float* ws  = (float*)d_ws;
  float* h   = ws;                                    // N*OUT floats
  float* acc = ws + (size_t)N_NODES * OUT_CH;         // N*OUT floats
  float* dnv = ws + 2 * (size_t)N_NODES * OUT_CH;     // N floats

  const int* src = edge_index;
  const int* dst = edge_index + E_EDGES;

  const int nThreads = (N_NODES + 255) / 256;
  const int feThreads = (N_NODES * OUT_CH + 255) / 256;

  k_deg_init <<<nThreads, 256, 0, stream>>>(dnv);
  k_deg_count<<<(E_EDGES + 255) / 256, 256, 0, stream>>>(dst, dnv);
  k_dinv     <<<nThreads, 256, 0, stream>>>(dnv);

  // 100000 rows / (2 M-tiles of 16 per block) = 3125 blocks exactly
  k_gemm_relu<<<N_NODES / 32, 256, 0, stream>>>(x, W, b, h, z);

  for (int it = 0; it < K_STEPS; ++it) {
    k_zero   <<<feThreads, 256, 0, stream>>>(acc);
    k_scatter<<<E_EDGES / 8, 256, 0, stream>>>(src, dst, dnv, z, acc);
    k_combine<<<feThreads, 256, 0, stream>>>(acc, h, dnv, z);
  }

  k_logsoftmax<<<(N_NODES + 7) / 8, 256, 0, stream>>>(z);
}